// MultiLoss_81990925680773
// MI455X (gfx1250) — compile-verified
//
#include <hip/hip_runtime.h>
#include <math.h>

// ---------------------------------------------------------------------------
// MultiLoss for MI455X (gfx1250).
// Memory roofline: ~9MB ch0 plane + ~2.26MB mask => ~0.5us @ 23.3TB/s; the
// workload is HBM/launch bound, pure f32 VALU (no matmul => WMMA N/A).
// CDNA5-specific path: Tensor Data Mover (tensor_load_to_lds) double-buffered
// LDS staging in the background-BCE reduction, synchronized with the split
// TENSORcnt counter (s_wait_tensorcnt). Wave-0 issues the DMA; 8 waves/block
// consume from LDS with vectorized ds reads.
// ---------------------------------------------------------------------------

#define B_     16
#define C_     9
#define MAXX_  400
#define MAXY_  352
#define NTGT   48
#define PLANE  (MAXX_ * MAXY_)        /* 140800 */
#define ZSIZE_ 4.0f
#define VALIDIOU_ 0.3f

#define TILE     2048                          /* floats per TDM tile (8KB) */
#define GBX      23                            /* blocks per plane          */
#define NT_PLANE ((PLANE + TILE - 1) / TILE)   /* 69 = 23*3                 */
#define MSTRIDE  (NT_PLANE * TILE)             /* padded mask plane stride  */

typedef unsigned int uint32x4 __attribute__((ext_vector_type(4)));
typedef int          int32x4  __attribute__((ext_vector_type(4)));
typedef int          int32x8  __attribute__((ext_vector_type(8)));

// ROCm 7.2 (clang-22): 5-arg tensor_load_to_lds builtin.  amdgpu-toolchain
// (clang-23 + therock-10.0 headers shipping amd_gfx1250_TDM.h): 6-arg form.
#if __has_include(<hip/amd_detail/amd_gfx1250_TDM.h>)
#define TDM_LOAD(g0, g1, z4a, z4b) \
    __builtin_amdgcn_tensor_load_to_lds(g0, g1, z4a, z4b, (int32x8)(0), 0)
#else
#define TDM_LOAD(g0, g1, z4a, z4b) \
    __builtin_amdgcn_tensor_load_to_lds(g0, g1, z4a, z4b, 0)
#endif

// Build a 1-D TDM descriptor (CDNA5 ISA) and issue the DMA.
// tile_dim0 = TILE elements; tensor_dim0 = remaining elements, so the TDM's
// OOB zero-fill handles the ragged plane tail in LDS for us.
__device__ inline void tdm_issue_1d(const float* gsrc, unsigned lds_byte_off,
                                    unsigned remaining_elems)
{
#if defined(__HIP_DEVICE_COMPILE__)
    unsigned long long ga = (unsigned long long)(const void*)gsrc;
    uint32x4 g0;
    g0.x = 1u;                                            // count=1, user mode
    g0.y = lds_byte_off;                                  // lds_addr
    g0.z = (unsigned)(ga & 0xFFFFFFFFull);                // global_addr[31:0]
    g0.w = ((unsigned)(ga >> 32) & 0x01FFFFFFu)           // global_addr[56:32]
           | 0x80000000u;                                 // type=2 ("image")
    int32x8 g1;
    g1[0] = 0x00020000;                                   // data_size=2 (4B)
    g1[1] = (int)((remaining_elems & 0xFFFFu) << 16);     // tensor_dim0 lo16
    g1[2] = (int)((remaining_elems >> 16) | (1u << 16));  // dim0 hi16 | dim1=1
    g1[3] = (int)((unsigned)TILE << 16);                  // tile_dim0
    g1[4] = 0;                                            // tile_dim1/2 unused
    g1[5] = TILE;                                         // dim0_stride lo32
    g1[6] = 0;
    g1[7] = 0;
    int32x4 z4 = {0, 0, 0, 0};
    TDM_LOAD(g0, g1, z4, z4);
#else
    (void)gsrc; (void)lds_byte_off; (void)remaining_elems;
#endif
}

__device__ inline void wait_tensorcnt_le1()
{
#if defined(__HIP_DEVICE_COMPILE__)
    __builtin_amdgcn_s_wait_tensorcnt(1);
#endif
}
__device__ inline void wait_tensorcnt_0()
{
#if defined(__HIP_DEVICE_COMPILE__)
    __builtin_amdgcn_s_wait_tensorcnt(0);
#endif
}

__device__ inline float sigmoidf_(float x) { return 1.0f / (1.0f + expf(-x)); }

// ------------------------- rotated-box 3D IoU ------------------------------
__device__ inline void bev_corners(const float* bx, float cx[4], float cy[4])
{
    float x = bx[0], y = bx[1], w = bx[3], l = bx[4], th = bx[6];
    float c = cosf(th), s = sinf(th);
    const float sx[4] = {1.f, -1.f, -1.f, 1.f};
    const float sy[4] = {1.f, 1.f, -1.f, -1.f};
#pragma unroll
    for (int i = 0; i < 4; ++i) {
        float lx = sx[i] * w * 0.5f, ly = sy[i] * l * 0.5f;
        cx[i] = x + c * lx - s * ly;
        cy[i] = y + s * lx + c * ly;
    }
}

__device__ float iou3d_pair(const float* a, const float* b)
{
    float cax[4], cay[4], cbx[4], cby[4];
    bev_corners(a, cax, cay);
    bev_corners(b, cbx, cby);

    float vx[8], vy[8];
#pragma unroll
    for (int i = 0; i < 4; ++i) { vx[i] = cax[i]; vy[i] = cay[i]; }
#pragma unroll
    for (int i = 4; i < 8; ++i) { vx[i] = 0.f; vy[i] = 0.f; }
    int n = 4;

    // Sutherland-Hodgman clip by the 4 half-planes of box B (d >= 0 inside).
#pragma unroll
    for (int j = 0; j < 4; ++j) {
        float px = cbx[j], py = cby[j];
        float qx = cbx[(j + 1) & 3], qy = cby[(j + 1) & 3];
        float ea = -(qy - py);
        float eb = qx - px;
        float ec = -(ea * px + eb * py);
        float nvx[8], nvy[8];
        int m = 0;
        for (int k = 0; k < n; ++k) {
            int kn = (k + 1 < n) ? k + 1 : 0;
            float d0 = ea * vx[k] + eb * vy[k] + ec;
            float d1 = ea * vx[kn] + eb * vy[kn] + ec;
            if ((d0 >= 0.f) != (d1 >= 0.f)) {
                float den = d0 - d1;
                float t = d0 / (fabsf(den) > 1e-9f ? den : 1.0f);
                if (m < 8) {
                    nvx[m] = vx[k] + t * (vx[kn] - vx[k]);
                    nvy[m] = vy[k] + t * (vy[kn] - vy[k]);
                    ++m;
                }
            }
            if (d1 >= 0.f && m < 8) { nvx[m] = vx[kn]; nvy[m] = vy[kn]; ++m; }
        }
        n = m;
        for (int k = 0; k < n; ++k) { vx[k] = nvx[k]; vy[k] = nvy[k]; }
    }

    float s = 0.f;
    for (int k = 0; k < n; ++k) {
        int kn = (k + 1 < n) ? k + 1 : 0;
        s += vx[k] * vy[kn] - vx[kn] * vy[k];
    }
    float inter_bev = 0.5f * fabsf(s);
    float zhi = fminf(a[2] + a[5] * 0.5f, b[2] + b[5] * 0.5f);
    float zlo = fmaxf(a[2] - a[5] * 0.5f, b[2] - b[5] * 0.5f);
    float iv = inter_bev * fmaxf(zhi - zlo, 0.f);
    float va = a[3] * a[4] * a[5];
    float vb = b[3] * b[4] * b[5];
    return iv / (va + vb - iv + 1e-8f);
}

// --------------------------- kernel 0: init --------------------------------
// Mask planes are padded to the TDM tile grid (MSTRIDE); pad bytes get 0 so
// the hot loop needs no range predicate.
__global__ __launch_bounds__(256) void k_init(unsigned int* __restrict__ mask32,
                                              float* __restrict__ accum)
{
    int i = blockIdx.x * 256 + threadIdx.x;
    if (i < (B_ * MSTRIDE) / 4) {
        int local = i % (MSTRIDE / 4);             // dword index within plane
        mask32[i] = (local < PLANE / 4) ? 0x01010101u : 0u;
    }
    if (i < 2 * B_) accum[i] = 0.0f;
}

// ------------------- kernel 1: IoU neighbor mask clearing ------------------
__global__ __launch_bounds__(256) void k_mask_clear(
    const float* __restrict__ out, const float* __restrict__ tgt,
    unsigned char* __restrict__ mask)
{
    int tid = blockIdx.x * 256 + threadIdx.x;
    if (tid >= B_ * NTGT * 8) return;
    int j = tid & 7;
    int rest = tid >> 3;
    int ti = rest % NTGT;
    int b = rest / NTGT;

    const float* t = tgt + (size_t)(b * NTGT + ti) * 7;
    float tb[7];
#pragma unroll
    for (int k = 0; k < 7; ++k) tb[k] = t[k];
    int dcx = (int)floorf(tb[0]);
    int dcy = (int)floorf(tb[1]);
    if (j == 0) mask[(size_t)b * MSTRIDE + dcx * MAXY_ + dcy] = 0;

    const int dxs[8] = {-1, 0, 1, -1, 1, -1, 0, 1};
    const int dys[8] = {-1, -1, -1, 0, 0, 1, 1, 1};
    int nx = dcx + dxs[j], ny = dcy + dys[j];
    if (nx < 0 || nx >= MAXX_ || ny < 0 || ny >= MAXY_) return;

    size_t cell = (size_t)b * C_ * PLANE + (size_t)nx * MAXY_ + ny;
    float r1 = out[cell + 1 * PLANE], r2 = out[cell + 2 * PLANE];
    float r3 = out[cell + 3 * PLANE], r4 = out[cell + 4 * PLANE];
    float r5 = out[cell + 5 * PLANE], r6 = out[cell + 6 * PLANE];
    float r7 = out[cell + 7 * PLANE], r8 = out[cell + 8 * PLANE];

    float pb[7];
    pb[0] = sigmoidf_(r1) + (float)nx;
    pb[1] = sigmoidf_(r2) + (float)ny;
    pb[2] = sigmoidf_(r3) * ZSIZE_;
    pb[3] = expf(r4) * 1.6f;
    pb[4] = expf(r5) * 3.9f;
    pb[5] = expf(r6) * 1.56f;
    pb[6] = atan2f(tanhf(r7), tanhf(r8));

    float iou = iou3d_pair(pb, tb);
    if (iou > VALIDIOU_) mask[(size_t)b * MSTRIDE + nx * MAXY_ + ny] = 0;
}

// ----------------- kernel 2: background BCE reduction (TDM) ----------------
__global__ __launch_bounds__(256) void k_bkg(const float* __restrict__ out,
                                             const unsigned char* __restrict__ mask,
                                             float* __restrict__ accum)
{
    __shared__ float sbuf[2][TILE];
    __shared__ float red0[256];
    __shared__ float red1[256];
    const int tid = threadIdx.x;
    const int b = blockIdx.y;
    const int g = blockIdx.x;
    const float* plane = out + (size_t)b * C_ * PLANE;    // channel-0 plane
    const unsigned char* mplane = mask + (size_t)b * MSTRIDE;

    const bool leader = (tid < 32);                       // wave 0 (wave32)
    if (leader) {
        tdm_issue_1d(plane + (size_t)g * TILE, (unsigned)(size_t)&sbuf[0][0],
                     (unsigned)(PLANE - g * TILE));
    }
    float s_bce = 0.f, s_cnt = 0.f;

#pragma unroll 1
    for (int it = 0; it < 3; ++it) {
        int t = g + it * GBX;                              // tile index < 69
        int buf = it & 1;
        if (leader) {
            if (it + 1 < 3) {
                int tn = g + (it + 1) * GBX;
                tdm_issue_1d(plane + (size_t)tn * TILE,
                             (unsigned)(size_t)&sbuf[(it + 1) & 1][0],
                             (unsigned)(PLANE - tn * TILE));
                wait_tensorcnt_le1();   // oldest (current tile) complete
            } else {
                wait_tensorcnt_0();
            }
        }
        __syncthreads();          // DMA-complete broadcast to all waves

        // 8 contiguous floats per thread from LDS + 8 mask bytes (padded, so
        // no range predicate; TDM zero-filled the f32 tail, pad mask is 0).
        const float4* sv = (const float4*)(&sbuf[buf][tid * 8]);
        float4 x0 = sv[0];
        float4 x1 = sv[1];
        unsigned long long mv =
            *(const unsigned long long*)(mplane + (size_t)t * TILE + tid * 8);
        float xs[8] = {x0.x, x0.y, x0.z, x0.w, x1.x, x1.y, x1.z, x1.w};
#pragma unroll
        for (int k = 0; k < 8; ++k) {
            float m = (float)((unsigned)(mv >> (8 * k)) & 0xFFu);
            float p = 1.0f / (1.0f + expf(-xs[k]));
            // -max(log1p(-p), -100) == min(softplus(x), 100)
            float bce = fminf(-log1pf(-p), 100.0f);
            s_bce += bce * m;
            s_cnt += m;
        }
        __syncthreads();          // buffer consumed before reuse at it+2
    }

    red0[tid] = s_bce;
    red1[tid] = s_cnt;
    __syncthreads();
    for (int s = 128; s > 0; s >>= 1) {
        if (tid < s) { red0[tid] += red0[tid + s]; red1[tid] += red1[tid + s]; }
        __syncthreads();
    }
    if (tid == 0) {
        atomicAdd(&accum[b], red0[0]);
        atomicAdd(&accum[B_ + b], red1[0]);
    }
}

// ----------------- kernel 3: per-target losses + final scalar --------------
__global__ __launch_bounds__(768) void k_final(const float* __restrict__ out,
                                               const float* __restrict__ tgt,
                                               const float* __restrict__ accum,
                                               float* __restrict__ dst)
{
    __shared__ float red[768];
    const int tid = threadIdx.x;
    const int b = tid / NTGT;
    const int i = tid % NTGT;

    const float* t = tgt + (size_t)(b * NTGT + i) * 7;
    float t0 = t[0], t1 = t[1], t2 = t[2], t6 = t[6];
    int dcx = (int)floorf(t0), dcy = (int)floorf(t1);
    size_t cell = (size_t)b * C_ * PLANE + (size_t)dcx * MAXY_ + dcy;
    float p0 = sigmoidf_(out[cell]);
    float p1 = sigmoidf_(out[cell + 1 * PLANE]);
    float p2 = sigmoidf_(out[cell + 2 * PLANE]);
    float p3 = sigmoidf_(out[cell + 3 * PLANE]);
    float o7 = tanhf(out[cell + 7 * PLANE]);
    float o8 = tanhf(out[cell + 8 * PLANE]);

    const float TWO_PI = 6.283185307179586f;
    float tt[3] = {t0 - floorf(t0), t1 - floorf(t1), t2 / ZSIZE_};
    float s7 = sinf(TWO_PI - t6);
    float c8 = cosf(TWO_PI - t6);

    float cls = fminf(-logf(p0), 100.0f) * (1.0f / NTGT);
    float ps[3] = {p1, p2, p3};
    float xyz = 0.f, wlh = 0.f;
#pragma unroll
    for (int k = 0; k < 3; ++k) {
        float lp = fmaxf(logf(ps[k]), -100.0f);
        float lq = fmaxf(log1pf(-ps[k]), -100.0f);
        xyz += -(tt[k] * lp + (1.0f - tt[k]) * lq);
        float d = ps[k] - tt[k];
        wlh += d * d;
    }
    xyz *= 1.0f / (NTGT * 3);
    wlh *= 0.5f / (NTGT * 3);
    float dim = o7 - s7, dre = o8 - c8;
    float lim = dim * dim * (1.0f / NTGT);
    float lre = dre * dre * (1.0f / NTGT);
    float nrm = 1.0f - sqrtf(o7 * o7 + o8 * o8);
    float limre = nrm * nrm * (1.0f / NTGT);

    red[tid] = cls + xyz + wlh + lim + lre + limre;
    __syncthreads();
    if (tid < 256) red[tid] += red[tid + 256] + red[tid + 512];
    __syncthreads();
    for (int s = 128; s > 0; s >>= 1) {
        if (tid < s) red[tid] += red[tid + s];
        __syncthreads();
    }
    if (tid == 0) {
        float bkg = 0.f;
#pragma unroll
        for (int bb = 0; bb < B_; ++bb) bkg += accum[bb] / accum[B_ + bb];
        dst[0] = red[0] + bkg;
    }
}

// ---------------------------------------------------------------------------
extern "C" void kernel_launch(void* const* d_in, const int* in_sizes, int n_in,
                              void* d_out, int out_size, void* d_ws, size_t ws_size,
                              hipStream_t stream)
{
    (void)in_sizes; (void)n_in; (void)out_size; (void)ws_size;
    const float* out = (const float*)d_in[0];   // (16,9,400,352) f32
    const float* tgt = (const float*)d_in[1];   // (16,48,7) f32
    float* accum = (float*)d_ws;                            // 32 floats
    unsigned char* mask = (unsigned char*)d_ws + 128;       // 16*MSTRIDE bytes

    k_init<<<(B_ * MSTRIDE / 4 + 255) / 256, 256, 0, stream>>>((unsigned int*)mask, accum);
    k_mask_clear<<<(B_ * NTGT * 8 + 255) / 256, 256, 0, stream>>>(out, tgt, mask);
    k_bkg<<<dim3(GBX, B_), 256, 0, stream>>>(out, mask, accum);
    k_final<<<1, 768, 0, stream>>>(out, tgt, accum, (float*)d_out);
}